// TurboTokenPermuter_73169062854664
// MI455X (gfx1250) — compile-verified
//
#include <hip/hip_runtime.h>
#include <hip/hip_bf16.h>

#define HIDDEN 4096
#define NEXP   64

typedef float v4f __attribute__((ext_vector_type(4)));
typedef int   v4i __attribute__((vector_size(16)));   // matches builtin param type

// -------- kernel 4 (placed first so disasm snippet shows it):
// gather rows via async global->LDS->global DMA copy.
// One 256-thread block per output row (16 KB). Each thread moves 4 x 16B.
// Pass stride = 256 threads * 16 B = 4096 B; 4 passes = 16384 B = 4096 floats.
__global__ void moe_gather_kernel(const float* __restrict__ tokens,
                                  const int* __restrict__ out_idx,
                                  float* __restrict__ out_tokens)
{
    const int row = blockIdx.x;
    const int tid = threadIdx.x;               // 0..255
    const int tok = out_idx[row];
    const float* src = tokens     + (size_t)tok * HIDDEN;
    float*       dst = out_tokens + (size_t)row * HIDDEN;

#if defined(__gfx1250__) &&                                                    \
    __has_builtin(__builtin_amdgcn_global_load_async_to_lds_b128) &&           \
    __has_builtin(__builtin_amdgcn_global_store_async_from_lds_b128)
    __shared__ alignas(16) unsigned char lds_buf[HIDDEN * 4];  // 16 KB

    v4i* g_src = (v4i*)((char*)src + (size_t)tid * 16);
    v4i* g_dst = (v4i*)((char*)dst + (size_t)tid * 16);
    v4i* l     = (v4i*)(&lds_buf[tid * 16]);

    // 4 async b128 loads per thread (RT hint: keep tokens L2-resident)
    __builtin_amdgcn_global_load_async_to_lds_b128(g_src, l, 0,     0);
    __builtin_amdgcn_global_load_async_to_lds_b128(g_src, l, 4096,  0);
    __builtin_amdgcn_global_load_async_to_lds_b128(g_src, l, 8192,  0);
    __builtin_amdgcn_global_load_async_to_lds_b128(g_src, l, 12288, 0);

#if __has_builtin(__builtin_amdgcn_s_wait_asynccnt)
    __builtin_amdgcn_s_wait_asynccnt(0);
#else
    asm volatile("s_wait_asynccnt 0" ::: "memory");
#endif

    // 4 async b128 stores per thread (TH=NT: don't rinse token table from L2)
    __builtin_amdgcn_global_store_async_from_lds_b128(g_dst, l, 0,     1);
    __builtin_amdgcn_global_store_async_from_lds_b128(g_dst, l, 4096,  1);
    __builtin_amdgcn_global_store_async_from_lds_b128(g_dst, l, 8192,  1);
    __builtin_amdgcn_global_store_async_from_lds_b128(g_dst, l, 12288, 1);
    // implicit wait-idle at s_endpgm covers outstanding async stores
#else
    // Fallback: vectorized copy with non-temporal stores.
    const v4f* s4 = (const v4f*)src;
    v4f*       d4 = (v4f*)dst;
#pragma unroll
    for (int p = 0; p < 4; ++p) {
        v4f v = s4[tid + p * 256];
        __builtin_nontemporal_store(v, &d4[tid + p * 256]);
    }
#endif
}

// ---------------- kernel 1: per-expert counts (one block per expert) --------
// Unrolled x4 with independent loads so the compiler batches them under one
// s_wait_loadcnt instead of a dependent load+wait per byte.
__global__ void moe_count_kernel(const unsigned char* __restrict__ map, int T,
                                 int* __restrict__ counts)
{
    const int e    = blockIdx.x;
    const int step = blockDim.x;               // 256
    int sum = 0;
    int t = threadIdx.x;
    for (; t + 3 * step < T; t += 4 * step) {
        const unsigned a0 = map[(size_t)(t)            * NEXP + e];
        const unsigned a1 = map[(size_t)(t +     step) * NEXP + e];
        const unsigned a2 = map[(size_t)(t + 2 * step) * NEXP + e];
        const unsigned a3 = map[(size_t)(t + 3 * step) * NEXP + e];
        sum += (a0 != 0) + (a1 != 0) + (a2 != 0) + (a3 != 0);
    }
    for (; t < T; t += step)
        sum += (map[(size_t)t * NEXP + e] != 0);

    __shared__ int red[256];
    red[threadIdx.x] = sum;
    __syncthreads();
    for (int off = 128; off > 0; off >>= 1) {
        if ((int)threadIdx.x < off) red[threadIdx.x] += red[threadIdx.x + off];
        __syncthreads();
    }
    if (threadIdx.x == 0) counts[e] = red[0];
}

// ---------------- kernel 2: exclusive scan over 64 experts ------------------
__global__ void moe_scan_kernel(const int* __restrict__ counts,
                                int* __restrict__ offsets)
{
    if (threadIdx.x == 0) {
        int acc = 0;
        for (int e = 0; e < NEXP; ++e) { offsets[e] = acc; acc += counts[e]; }
        offsets[NEXP] = acc;
    }
}

// -------- kernel 3: one wave32 per expert, stable intra-expert ranks --------
__global__ void moe_assign_kernel(const unsigned char* __restrict__ map,
                                  const float* __restrict__ probs, int T,
                                  const int* __restrict__ offsets,
                                  float* __restrict__ out_probs,
                                  int* __restrict__ out_idx)
{
    const int e    = blockIdx.x;
    const int lane = threadIdx.x;              // 0..31 (wave32)
    int base = offsets[e];
    for (int t0 = 0; t0 < T; t0 += 32) {
        const int  t = t0 + lane;
        const bool f = (t < T) && (map[(size_t)t * NEXP + e] != 0);
        const unsigned mask = __builtin_amdgcn_ballot_w32(f);
        const unsigned lt   = (lane == 0) ? 0u : (0xFFFFFFFFu >> (32 - lane));
        if (f) {
            const int row = base + __popc(mask & lt);
            out_idx[row]   = t;
            out_probs[row] = probs[(size_t)t * NEXP + e];
        }
        base += __popc(mask);                  // uniform (SGPR) update
    }
}

extern "C" void kernel_launch(void* const* d_in, const int* in_sizes, int n_in,
                              void* d_out, int out_size, void* d_ws, size_t ws_size,
                              hipStream_t stream)
{
    const float*         tokens = (const float*)d_in[0];
    const float*         probs  = (const float*)d_in[1];
    const unsigned char* map    = (const unsigned char*)d_in[2];  // bool[T,E]
    // d_in[3] = num_out_tokens (device scalar) — derived on host instead.

    const int T = in_sizes[0] / HIDDEN;                 // 8192
    const long long R = (long long)out_size / (HIDDEN + 2);  // 65536 out rows

    float* out_tokens = (float*)d_out;                  // [R, HIDDEN]
    float* out_probs  = out_tokens + (size_t)R * HIDDEN;// [R]
    int*   out_idx    = (int*)(out_probs + R);          // [R] int32 bits

    int* counts  = (int*)d_ws;                          // [64]
    int* offsets = counts + NEXP;                       // [65]

    moe_count_kernel <<<NEXP, 256, 0, stream>>>(map, T, counts);
    moe_scan_kernel  <<<1,    64,  0, stream>>>(counts, offsets);
    moe_assign_kernel<<<NEXP, 32,  0, stream>>>(map, probs, T, offsets,
                                                out_probs, out_idx);
    moe_gather_kernel<<<(int)R, 256, 0, stream>>>(tokens, out_idx, out_tokens);
}